// KBRD_48850958025294
// MI455X (gfx1250) — compile-verified
//
#include <hip/hip_runtime.h>
#include <hip/hip_bf16.h>
#include <math.h>

#define N_ENTITY 64368
#define N_REL    40
#define DIM      128
#define NUM_BASES 8
#define N_EDGES  1200000
#define BATCH    64
#define MAX_SEED 32

typedef __attribute__((ext_vector_type(2))) float v2f;
typedef __attribute__((ext_vector_type(8))) float v8f;
typedef __attribute__((ext_vector_type(4))) unsigned tdm_g0_t;
typedef __attribute__((ext_vector_type(8))) unsigned tdm_g1_t;

// ---------------- workspace layout (floats) ----------------
// nodes : [N_ENTITY * DIM]
// u     : [BATCH * DIM]
// stats : [BATCH * 2]  (row max, row sumexp)
#define WS_NODES 0
#define WS_U     (N_ENTITY * DIM)
#define WS_STATS (WS_U + BATCH * DIM)

// ---------------- kernel A: nodes = root + rgcn_bias ----------------
__global__ void k_init_nodes(const float* __restrict__ root,
                             const float* __restrict__ bias,
                             float* __restrict__ nodes) {
    int i = blockIdx.x * blockDim.x + threadIdx.x;       // float4 index
    const int total = N_ENTITY * DIM / 4;
    if (i >= total) return;
    const float4* r4 = (const float4*)root;
    const float4* b4 = (const float4*)bias;
    float4 v = r4[i];
    float4 b = b4[i & (DIM / 4 - 1)];
    v.x += b.x; v.y += b.y; v.z += b.z; v.w += b.w;
    ((float4*)nodes)[i] = v;
}

// ---------------- kernel B: edge messages, one wave per edge ----------------
// msg = sum_b att[rel][b] * basis[b][src]; atomicAdd into nodes[dst]
__global__ void k_edges(const float* __restrict__ basis,
                        const float* __restrict__ att,
                        const int*   __restrict__ edge_index,
                        const int*   __restrict__ edge_type,
                        float* __restrict__ nodes) {
    const int lane = threadIdx.x & 31;
    const int wave = threadIdx.x >> 5;
    const int e = blockIdx.x * 8 + wave;                 // 8 waves / block
    if (e >= N_EDGES) return;

    const int src = edge_index[e];
    const int dst = edge_index[N_EDGES + e];
    const int rel = edge_type[e];

    float c[NUM_BASES];
#pragma unroll
    for (int b = 0; b < NUM_BASES; ++b) c[b] = att[rel * NUM_BASES + b];

    float4 acc = make_float4(0.f, 0.f, 0.f, 0.f);
    const int d0 = lane * 4;                             // 32 lanes x 4 = 128 dims
#pragma unroll
    for (int b = 0; b < NUM_BASES; ++b) {
        const float4 v = *(const float4*)(basis +
            ((size_t)b * N_ENTITY + (size_t)src) * DIM + d0);
        acc.x += c[b] * v.x; acc.y += c[b] * v.y;
        acc.z += c[b] * v.z; acc.w += c[b] * v.w;
    }
    float* p = nodes + (size_t)dst * DIM + d0;
    atomicAdd(p + 0, acc.x);
    atomicAdd(p + 1, acc.y);
    atomicAdd(p + 2, acc.z);
    atomicAdd(p + 3, acc.w);
}

// ---------------- kernel C: per-batch attention pooling -> u ----------------
__global__ void k_attn_pool(const float* __restrict__ nodes,
                            const float* __restrict__ attn_a,   // [DIM, DIM]
                            const float* __restrict__ attn_b,   // [DIM]
                            const int*   __restrict__ seed_idx, // [BATCH, 32]
                            const unsigned char* __restrict__ seed_mask,
                            float* __restrict__ u) {
    __shared__ float hs[MAX_SEED * DIM];   // 16 KB
    __shared__ float e_s[MAX_SEED];

    const int b    = blockIdx.x;
    const int tid  = threadIdx.x;          // 128 threads = 4 waves
    const int lane = tid & 31;
    const int wave = tid >> 5;

    // gather h = nodes[seed_idx[b]]
    for (int l = 0; l < MAX_SEED; ++l) {
        const int idx = seed_idx[b * MAX_SEED + l];
        hs[l * DIM + tid] = nodes[(size_t)idx * DIM + tid];
    }
    __syncthreads();

    // e_l = tanh(h_l @ A) . attn_b  (wave handles l = wave, wave+4, ...)
    const float bv0 = attn_b[lane];
    const float bv1 = attn_b[lane + 32];
    const float bv2 = attn_b[lane + 64];
    const float bv3 = attn_b[lane + 96];
    for (int l = wave; l < MAX_SEED; l += 4) {
        float a0 = 0.f, a1 = 0.f, a2 = 0.f, a3 = 0.f;
        for (int k = 0; k < DIM; ++k) {
            const float hv = hs[l * DIM + k];
            const float* Ar = attn_a + k * DIM;
            a0 += hv * Ar[lane];
            a1 += hv * Ar[lane + 32];
            a2 += hv * Ar[lane + 64];
            a3 += hv * Ar[lane + 96];
        }
        float s = tanhf(a0) * bv0 + tanhf(a1) * bv1 +
                  tanhf(a2) * bv2 + tanhf(a3) * bv3;
#pragma unroll
        for (int off = 16; off > 0; off >>= 1) s += __shfl_xor(s, off, 32);
        if (lane == 0)
            e_s[l] = seed_mask[b * MAX_SEED + l] ? s : -INFINITY;
    }
    __syncthreads();

    // masked softmax over l, then u_d = sum_l attn_l * h[l][d]
    float m = -INFINITY;
#pragma unroll
    for (int l = 0; l < MAX_SEED; ++l) m = fmaxf(m, e_s[l]);
    float denom = 0.f;
#pragma unroll
    for (int l = 0; l < MAX_SEED; ++l) denom += __expf(e_s[l] - m);
    float ud = 0.f;
#pragma unroll
    for (int l = 0; l < MAX_SEED; ++l)
        ud += __expf(e_s[l] - m) * hs[l * DIM + tid];
    u[b * DIM + tid] = ud / denom;
}

// ---------------- TDM helper: 2D tile load (global -> LDS) --------------------
// Row = DIM f32 (512B). HW padding: every 128 DWORDs insert 2 DWORDs ->
// LDS row stride = 130 floats (bank-conflict-free WMMA fragment reads).
// OOB rows (tile rows beyond tensor_rows) read as zero by the TDM.
__device__ __forceinline__ void tdm_load_rows(unsigned lds_byte_addr,
                                              const float* gptr,
                                              unsigned tensor_rows,
                                              unsigned tile_rows) {
    const unsigned long long ga = (unsigned long long)(uintptr_t)gptr;
    tdm_g0_t g0;
    g0[0] = 1u;                                      // count=1 (valid), user mode
    g0[1] = lds_byte_addr;                           // lds_addr [63:32]
    g0[2] = (unsigned)ga;                            // global_addr [95:64]
    g0[3] = (unsigned)(ga >> 32) | (2u << 30);       // addr[56:32] | type=2
    tdm_g1_t g1;
    g1[0] = (2u << 16)        // data_size = 4B
          | (1u << 20)        // pad_enable
          | (6u << 22)        // pad_interval: every 128 DWORDs
          | (1u << 25);       // pad_amount: 2 DWORDs
    g1[1] = ((unsigned)DIM & 0xFFFFu) << 16;         // tensor_dim0 low16
    g1[2] = (tensor_rows & 0xFFFFu) << 16;           // dim0 hi=0 | tensor_dim1 lo
    g1[3] = ((tensor_rows >> 16) & 0xFFFFu)          // tensor_dim1 hi
          | ((unsigned)DIM << 16);                   // tile_dim0 = 128
    g1[4] = tile_rows & 0xFFFFu;                     // tile_dim1 | tile_dim2=0
    g1[5] = (unsigned)DIM;                           // tensor_dim0_stride low32
    g1[6] = 0u;
    g1[7] = 0u;
    asm volatile("tensor_load_to_lds %0, %1" :: "s"(g0), "s"(g1) : "memory");
}

// ---------------- kernel D: scores = u @ nodes^T + bias via f32 WMMA ----------
// Block: 256 threads = 8 waves. Covers M=64 (4 tiles) x N=32 (2 tiles).
#define LSTR 130   // padded LDS row stride (floats) -> conflict-free fragments
__global__ void k_scores_wmma(const float* __restrict__ u,      // [64,128]
                              const float* __restrict__ nodes,  // [N,128]
                              const float* __restrict__ outb,   // [N]
                              float* __restrict__ scores) {     // [64,N]
    __shared__ float u_s[BATCH * LSTR];     // ~33 KB (TDM-padded rows)
    __shared__ float nd_s[32 * LSTR];       // ~17 KB (TDM-padded rows)

    const int tid  = threadIdx.x;
    const int lane = tid & 31;
    const int wave = tid >> 5;
    const int n_base = blockIdx.x * 32;

    // Wave 0 issues both tile DMAs via the Tensor Data Mover, waits on
    // TENSORcnt, then the workgroup barrier publishes the LDS tiles.
    if (wave == 0) {
        tdm_load_rows((unsigned)(uintptr_t)&u_s[0], u,
                      (unsigned)BATCH, (unsigned)BATCH);
        tdm_load_rows((unsigned)(uintptr_t)&nd_s[0],
                      nodes + (size_t)n_base * DIM,
                      (unsigned)(N_ENTITY - n_base),   // OOB tile rows -> 0
                      32u);
        __builtin_amdgcn_s_wait_tensorcnt(0);
    }
    __syncthreads();

    const int mt = wave & 3;                    // M tile 0..3
    const int nt = wave >> 2;                   // N tile 0..1
    const int arow = mt * 16 + (lane & 15);
    const int bcol = nt * 16 + (lane & 15);
    const int koff = (lane >> 4) << 1;          // 0 or 2

    v8f acc = {};
#pragma unroll
    for (int k = 0; k < DIM; k += 4) {
        const v2f a  = *(const v2f*)&u_s [arow * LSTR + k + koff];
        const v2f bb = *(const v2f*)&nd_s[bcol * LSTR + k + koff];
        acc = __builtin_amdgcn_wmma_f32_16x16x4_f32(
                false, a, false, bb, (short)0, acc, false, false);
    }

    const int gcol = n_base + nt * 16 + (lane & 15);
    if (gcol < N_ENTITY) {
        const float bias = outb[gcol];
        const int mhalf = (lane >> 4) * 8;
#pragma unroll
        for (int r = 0; r < 8; ++r) {
            const int gm = mt * 16 + mhalf + r;
            scores[(size_t)gm * N_ENTITY + gcol] = acc[r] + bias;
        }
    }
}

// ---------------- kernel E: per-row max / sum-exp over scores ----------------
__global__ void k_row_stats(const float* __restrict__ scores,
                            float* __restrict__ stats) {
    __shared__ float red[256];
    const int b = blockIdx.x, tid = threadIdx.x;
    const float* row = scores + (size_t)b * N_ENTITY;

    float m = -INFINITY;
    for (int i = tid; i < N_ENTITY; i += 256) m = fmaxf(m, row[i]);
    red[tid] = m; __syncthreads();
    for (int s = 128; s > 0; s >>= 1) {
        if (tid < s) red[tid] = fmaxf(red[tid], red[tid + s]);
        __syncthreads();
    }
    m = red[0]; __syncthreads();

    float sum = 0.f;
    for (int i = tid; i < N_ENTITY; i += 256) sum += __expf(row[i] - m);
    red[tid] = sum; __syncthreads();
    for (int s = 128; s > 0; s >>= 1) {
        if (tid < s) red[tid] += red[tid + s];
        __syncthreads();
    }
    if (tid == 0) { stats[2 * b] = m; stats[2 * b + 1] = red[0]; }
}

// ---------------- kernel F: CE loss ----------------
__global__ void k_loss(const float* __restrict__ scores,
                       const float* __restrict__ stats,
                       const int* __restrict__ labels,
                       float* __restrict__ loss_out) {
    __shared__ float red[BATCH];
    const int b = threadIdx.x;
    const float m = stats[2 * b];
    const float z = stats[2 * b + 1];
    const float x = scores[(size_t)b * N_ENTITY + labels[b]];
    red[b] = x - m - logf(z);
    __syncthreads();
    for (int s = 32; s > 0; s >>= 1) {
        if (b < s) red[b] += red[b + s];
        __syncthreads();
    }
    if (b == 0) *loss_out = -red[0] / (float)BATCH;
}

extern "C" void kernel_launch(void* const* d_in, const int* in_sizes, int n_in,
                              void* d_out, int out_size, void* d_ws, size_t ws_size,
                              hipStream_t stream) {
    const float* basis     = (const float*)d_in[0];
    const float* att       = (const float*)d_in[1];
    const float* root      = (const float*)d_in[2];
    const float* rgcn_bias = (const float*)d_in[3];
    const float* attn_a    = (const float*)d_in[4];
    const float* attn_b    = (const float*)d_in[5];
    const float* outb      = (const float*)d_in[6];
    const int*   edge_idx  = (const int*)d_in[7];
    const int*   edge_type = (const int*)d_in[8];
    const int*   seed_idx  = (const int*)d_in[9];
    const unsigned char* seed_mask = (const unsigned char*)d_in[10];
    const int*   labels    = (const int*)d_in[11];

    float* scores = (float*)d_out;                          // [64, N_ENTITY]
    float* loss   = scores + (size_t)BATCH * N_ENTITY;      // scalar
    float* ws     = (float*)d_ws;
    float* nodes  = ws + WS_NODES;
    float* u      = ws + WS_U;
    float* stats  = ws + WS_STATS;

    // A: nodes = root + bias
    {
        const int total = N_ENTITY * DIM / 4;
        k_init_nodes<<<(total + 255) / 256, 256, 0, stream>>>(root, rgcn_bias, nodes);
    }
    // B: scatter-add edge messages (one wave per edge)
    k_edges<<<(N_EDGES + 7) / 8, 256, 0, stream>>>(basis, att, edge_idx, edge_type, nodes);
    // C: attention pooling -> u
    k_attn_pool<<<BATCH, DIM, 0, stream>>>(nodes, attn_a, attn_b, seed_idx, seed_mask, u);
    // D: scores = u @ nodes^T + bias  (TDM tile staging + f32 WMMA)
    k_scores_wmma<<<(N_ENTITY + 31) / 32, 256, 0, stream>>>(u, nodes, outb, scores);
    // E: per-row log-softmax stats
    k_row_stats<<<BATCH, 256, 0, stream>>>(scores, stats);
    // F: loss
    k_loss<<<1, BATCH, 0, stream>>>(scores, stats, labels, loss);
}